// GINDEBUG_90726889161564
// MI455X (gfx1250) — compile-verified
//
#include <hip/hip_runtime.h>

#define N_NODES 100000
#define N_EDGES 3200000
#define D_FEAT  128
#define HID     32
#define D_TGT   10
#define BN_EPS  1e-5f

typedef __attribute__((ext_vector_type(2))) float v2f;
typedef __attribute__((ext_vector_type(8))) float v8f;

static __device__ __forceinline__ v8f wmma_f32(v2f a, v2f b, v8f c) {
  // V_WMMA_F32_16X16X4_F32 : D = A(16x4) * B(4x16) + C(16x16), fp32 exact
  return __builtin_amdgcn_wmma_f32_16x16x4_f32(false, a, false, b, (short)0, c,
                                               false, false);
}

// ---------------------------------------------------------------------------
// Edge scatter: one wave32 per edge. Lane l loads float4 of src row feature
// chunk (coalesced 512B/wave) and atomically adds into dst row.
// ---------------------------------------------------------------------------
__global__ void __launch_bounds__(256) gin_scatter(const float* __restrict__ x0,
                                                   const long long* __restrict__ ei,
                                                   float* __restrict__ agg) {
  unsigned gt   = blockIdx.x * 256u + threadIdx.x;
  unsigned e    = gt >> 5;
  unsigned lane = gt & 31u;
  if (e >= N_EDGES) return;
  long long s = ei[e];                       // edge_index[0][e]
  long long d = ei[(size_t)N_EDGES + e];     // edge_index[1][e]
  float4 v = *((const float4*)(x0 + (size_t)s * D_FEAT) + lane);
  float* ap = agg + (size_t)d * D_FEAT + lane * 4u;
  atomicAdd(ap + 0, v.x);
  atomicAdd(ap + 1, v.y);
  atomicAdd(ap + 2, v.z);
  atomicAdd(ap + 3, v.w);
}

// ---------------------------------------------------------------------------
// GEMM1: y1 = (x0 + agg) @ W1 + b1 over 16-row tiles, K=128 in steps of 4.
// Two 16x16 N-tiles cover HID=32. Epilogue stores pre-BN activations and
// accumulates per-column sum / sum-of-squares for BN statistics.
// ---------------------------------------------------------------------------
__global__ void __launch_bounds__(256) gin_gemm1(const float* __restrict__ x0,
                                                 const float* __restrict__ agg,
                                                 const float* __restrict__ W1,
                                                 const float* __restrict__ b1,
                                                 float* __restrict__ y1,
                                                 float* __restrict__ stats) {
  int tile = blockIdx.x * 8 + (threadIdx.x >> 5);
  if (tile >= N_NODES / 16) return;          // wave-uniform exit, EXEC stays full
  int lane = threadIdx.x & 31;
  int m    = lane & 15;                      // A row / B,D column
  int kh   = lane >> 4;                      // K-pair select (0 or 1)
  int row  = tile * 16 + m;
  const float* xr = x0  + (size_t)row * D_FEAT;
  const float* ar = agg + (size_t)row * D_FEAT;

  v8f acc0 = {}; v8f acc1 = {};
  #pragma unroll 4
  for (int k0 = 0; k0 < D_FEAT; k0 += 4) {
    int ka = k0 + kh * 2;
    v2f xa = *(const v2f*)(xr + ka);
    v2f aa = *(const v2f*)(ar + ka);
    v2f a;  a.x = xa.x + aa.x;  a.y = xa.y + aa.y;     // h = x0 + agg
    v2f bA, bB;
    bA.x = W1[ka * HID + m];        bA.y = W1[(ka + 1) * HID + m];
    bB.x = W1[ka * HID + 16 + m];   bB.y = W1[(ka + 1) * HID + 16 + m];
    acc0 = wmma_f32(a, bA, acc0);
    acc1 = wmma_f32(a, bB, acc1);
  }

  float bias0 = b1[m], bias1 = b1[m + 16];
  int   rbase = tile * 16 + kh * 8;          // D layout: vgpr j -> row j (+8 hi half)
  float s0 = 0.f, q0 = 0.f, s1 = 0.f, q1 = 0.f;
  #pragma unroll
  for (int j = 0; j < 8; ++j) {
    float v0 = acc0[j] + bias0;
    float v1 = acc1[j] + bias1;
    y1[(size_t)(rbase + j) * HID + m]      = v0;
    y1[(size_t)(rbase + j) * HID + 16 + m] = v1;
    s0 += v0; q0 += v0 * v0; s1 += v1; q1 += v1 * v1;
  }
  // lanes L and L+16 hold the same column -> fold before atomics
  s0 += __shfl_down(s0, 16, 32);  q0 += __shfl_down(q0, 16, 32);
  s1 += __shfl_down(s1, 16, 32);  q1 += __shfl_down(q1, 16, 32);
  if (lane < 16) {
    atomicAdd(&stats[m],          s0);   // sum, cols 0..15
    atomicAdd(&stats[32 + m],     q0);   // sumsq
    atomicAdd(&stats[16 + m],     s1);   // sum, cols 16..31
    atomicAdd(&stats[48 + m],     q1);
  }
}

// ---------------------------------------------------------------------------
// Fold BN stats + gamma/beta into per-column scale/shift: h = relu(y*sc + sh)
// ---------------------------------------------------------------------------
__global__ void gin_finalize(const float* __restrict__ raw,
                             const float* __restrict__ gamma,
                             const float* __restrict__ beta,
                             float* __restrict__ bn) {
  int c = threadIdx.x;
  if (c < HID) {
    const float invn = 1.0f / (float)N_NODES;
    float mean = raw[c] * invn;
    float var  = raw[32 + c] * invn - mean * mean;
    float sc   = rsqrtf(var + BN_EPS) * gamma[c];
    bn[c]      = sc;
    bn[32 + c] = fmaf(-mean, sc, beta[c]);
  }
}

// ---------------------------------------------------------------------------
// GEMM2: y2 = relu(bn1(y1)) @ W2 + b2, K=32. BN+ReLU fused into A-load.
// ---------------------------------------------------------------------------
__global__ void __launch_bounds__(256) gin_gemm_mid(const float* __restrict__ yin,
                                                    const float* __restrict__ bn,
                                                    const float* __restrict__ W2,
                                                    const float* __restrict__ b2,
                                                    float* __restrict__ yout,
                                                    float* __restrict__ stats) {
  int tile = blockIdx.x * 8 + (threadIdx.x >> 5);
  if (tile >= N_NODES / 16) return;
  int lane = threadIdx.x & 31;
  int m    = lane & 15;
  int kh   = lane >> 4;
  int row  = tile * 16 + m;
  const float* yr = yin + (size_t)row * HID;

  v8f acc0 = {}; v8f acc1 = {};
  #pragma unroll
  for (int k0 = 0; k0 < HID; k0 += 4) {
    int ka = k0 + kh * 2;
    v2f yv = *(const v2f*)(yr + ka);
    v2f a;
    a.x = fmaxf(fmaf(yv.x, bn[ka],     bn[32 + ka]),     0.f);
    a.y = fmaxf(fmaf(yv.y, bn[ka + 1], bn[32 + ka + 1]), 0.f);
    v2f bA, bB;
    bA.x = W2[ka * HID + m];        bA.y = W2[(ka + 1) * HID + m];
    bB.x = W2[ka * HID + 16 + m];   bB.y = W2[(ka + 1) * HID + 16 + m];
    acc0 = wmma_f32(a, bA, acc0);
    acc1 = wmma_f32(a, bB, acc1);
  }

  float bias0 = b2[m], bias1 = b2[m + 16];
  int   rbase = tile * 16 + kh * 8;
  float s0 = 0.f, q0 = 0.f, s1 = 0.f, q1 = 0.f;
  #pragma unroll
  for (int j = 0; j < 8; ++j) {
    float v0 = acc0[j] + bias0;
    float v1 = acc1[j] + bias1;
    yout[(size_t)(rbase + j) * HID + m]      = v0;
    yout[(size_t)(rbase + j) * HID + 16 + m] = v1;
    s0 += v0; q0 += v0 * v0; s1 += v1; q1 += v1 * v1;
  }
  s0 += __shfl_down(s0, 16, 32);  q0 += __shfl_down(q0, 16, 32);
  s1 += __shfl_down(s1, 16, 32);  q1 += __shfl_down(q1, 16, 32);
  if (lane < 16) {
    atomicAdd(&stats[m],      s0);
    atomicAdd(&stats[32 + m], q0);
    atomicAdd(&stats[16 + m], s1);
    atomicAdd(&stats[48 + m], q1);
  }
}

// ---------------------------------------------------------------------------
// Output: out = relu(bn2(y2)) @ W3 + b3, K=32, N=10 (B zero-padded to 16).
// ---------------------------------------------------------------------------
__global__ void __launch_bounds__(256) gin_out(const float* __restrict__ yin,
                                               const float* __restrict__ bn,
                                               const float* __restrict__ W3,
                                               const float* __restrict__ b3,
                                               float* __restrict__ out) {
  int tile = blockIdx.x * 8 + (threadIdx.x >> 5);
  if (tile >= N_NODES / 16) return;
  int lane = threadIdx.x & 31;
  int m    = lane & 15;
  int kh   = lane >> 4;
  int row  = tile * 16 + m;
  const float* yr = yin + (size_t)row * HID;

  v8f acc = {};
  #pragma unroll
  for (int k0 = 0; k0 < HID; k0 += 4) {
    int ka = k0 + kh * 2;
    v2f yv = *(const v2f*)(yr + ka);
    v2f a;
    a.x = fmaxf(fmaf(yv.x, bn[ka],     bn[32 + ka]),     0.f);
    a.y = fmaxf(fmaf(yv.y, bn[ka + 1], bn[32 + ka + 1]), 0.f);
    v2f b;
    b.x = (m < D_TGT) ? W3[ka * D_TGT + m]       : 0.f;   // select, not branch
    b.y = (m < D_TGT) ? W3[(ka + 1) * D_TGT + m] : 0.f;
    acc = wmma_f32(a, b, acc);
  }
  if (m < D_TGT) {                           // predicated stores only after WMMA
    float bias = b3[m];
    int   rbase = tile * 16 + kh * 8;
    #pragma unroll
    for (int j = 0; j < 8; ++j)
      out[(size_t)(rbase + j) * D_TGT + m] = acc[j] + bias;
  }
}

// ---------------------------------------------------------------------------
extern "C" void kernel_launch(void* const* d_in, const int* in_sizes, int n_in,
                              void* d_out, int out_size, void* d_ws, size_t ws_size,
                              hipStream_t stream) {
  const float*     x   = (const float*)d_in[0];
  const long long* ei  = (const long long*)d_in[1];   // int64 edge_index (2,E)
  // d_in[2] = edge_attr (unused by reference)
  const float* W1  = (const float*)d_in[3];
  const float* b1  = (const float*)d_in[4];
  const float* g1  = (const float*)d_in[5];
  const float* be1 = (const float*)d_in[6];
  const float* W2  = (const float*)d_in[7];
  const float* b2  = (const float*)d_in[8];
  const float* g2  = (const float*)d_in[9];
  const float* be2 = (const float*)d_in[10];
  const float* W3  = (const float*)d_in[11];
  const float* b3  = (const float*)d_in[12];
  float* out = (float*)d_out;
  float* ws  = (float*)d_ws;

  // workspace layout (floats)
  float* agg = ws;                                    // N*128
  float* raw = ws + (size_t)N_NODES * D_FEAT;         // 128: sum1,ss1,sum2,ss2
  float* fin = raw + 128;                             // 128: sc1,sh1,sc2,sh2
  float* y1  = fin + 128;                             // N*32
  float* y2  = y1 + (size_t)N_NODES * HID;            // N*32

  // zero agg + raw stats each call (atomically accumulated)
  hipMemsetAsync(agg, 0, ((size_t)N_NODES * D_FEAT + 128) * sizeof(float), stream);

  // scatter: one wave per edge
  {
    unsigned total = (unsigned)N_EDGES * 32u;         // 102.4M threads
    gin_scatter<<<dim3((total + 255u) / 256u), 256, 0, stream>>>(x, ei, agg);
  }

  const int tiles   = N_NODES / 16;                   // 6250 (exact)
  const int gblocks = (tiles + 7) / 8;                // 8 waves/block

  gin_gemm1   <<<gblocks, 256, 0, stream>>>(x, agg, W1, b1, y1, raw);
  gin_finalize<<<1, 32, 0, stream>>>(raw,      g1, be1, fin);
  gin_gemm_mid<<<gblocks, 256, 0, stream>>>(y1, fin,      W2, b2, y2, raw + 64);
  gin_finalize<<<1, 32, 0, stream>>>(raw + 64, g2, be2, fin + 64);
  gin_out     <<<gblocks, 256, 0, stream>>>(y2, fin + 64, W3, b3, out);
}